// MyModel_78606491452481
// MI455X (gfx1250) — compile-verified
//
#include <hip/hip_runtime.h>

// ---------------------------------------------------------------------------
// SAGPool GNN for MI455X (gfx1250, wave32). Dense GEMMs use full-precision
// V_WMMA_F32_16X16X4_F32 (fp32 A/B/C -> no precision loss vs fp32 reference).
// Sparse GCN aggregation uses native global_atomic_add_f32. Top-k = in-LDS
// bitonic sort per graph. LSTM recurrence = single-block WMMA with LDS state.
// ---------------------------------------------------------------------------

static constexpr int HDIM   = 256;       // hidden channels
static constexpr int NGRAPH = 256;       // graphs
static constexpr int PER1   = 512;       // nodes/graph before pool1
static constexpr int PER2   = 256;       // nodes/graph before pool2
static constexpr int KEEP1  = 256;       // k after pool1
static constexpr int KEEP2  = 128;       // k after pool2
static constexpr int NODE1  = NGRAPH * PER1;   // 131072
static constexpr int NODE2  = NGRAPH * KEEP1;  // 65536
static constexpr int NODE3  = NGRAPH * KEEP2;  // 32768
static constexpr int NEDGE  = NGRAPH * 4096;   // 1048576
static constexpr int INDIM  = 6;
static constexpr int NBL    = 4;

typedef __attribute__((ext_vector_type(2))) float v2f;
typedef __attribute__((ext_vector_type(8))) float v8f;

__device__ __forceinline__ float fatomic_add(float* p, float v) {
  return __hip_atomic_fetch_add(p, v, __ATOMIC_RELAXED, __HIP_MEMORY_SCOPE_AGENT);
}
__device__ __forceinline__ float sigmoidf(float x) { return 1.f / (1.f + expf(-x)); }

// ---------------------------------------------------------------------------
// GEMM: C[M x 256] = A[M x 256] * W[256 x 256] (+bias) (+relu)
// Block = 256 threads = 8 waves; block tile = 128(M) x 16(N); W panel in LDS.
// Wave tile = 16x16 via 64 chained V_WMMA_F32_16X16X4_F32.
// ---------------------------------------------------------------------------
__global__ __launch_bounds__(256) void gemm_wmma256(
    const float* __restrict__ A, const float* __restrict__ W,
    float* __restrict__ C, const float* __restrict__ bias, int relu) {
  __shared__ float wpan[4096];           // [(k>>1)][n][k&1] : ((k>>1)*16+n)*2+(k&1)
  const int tid = threadIdx.x;
  const int n0 = blockIdx.x * 16;
  {
    const int k = tid;                   // one W row per thread
    const float* wr = W + (size_t)k * 256 + n0;
    const int base = (k >> 1) * 32 + (k & 1);
#pragma unroll
    for (int n = 0; n < 16; ++n) wpan[base + n * 2] = wr[n];
  }
  __syncthreads();

  const int lane = tid & 31;
  const int col  = lane & 15;            // A: M row | B/D: N col (same lane value)
  const int kh   = lane >> 4;            // K-pair half / D M-half
  const int m0   = blockIdx.y * 128 + (tid >> 5) * 16;

  const float* arow = A + (size_t)(m0 + col) * 256 + 2 * kh;
  v8f acc;
  {
    const float bv = bias ? bias[n0 + col] : 0.f;
#pragma unroll
    for (int r = 0; r < 8; ++r) acc[r] = bv;
  }
#pragma unroll 8
  for (int k0 = 0; k0 < 256; k0 += 4) {
    v2f a = *(const v2f*)(arow + k0);
    v2f b = *(const v2f*)(&wpan[(k0 >> 1) * 32 + kh * 32 + col * 2]);
    acc = __builtin_amdgcn_wmma_f32_16x16x4_f32(false, a, false, b,
                                                (short)0, acc, false, false);
  }
  float* crow = C + (size_t)m0 * 256 + n0 + col;
#pragma unroll
  for (int r = 0; r < 8; ++r) {
    float v = acc[r];
    if (relu) v = fmaxf(v, 0.f);
    crow[(size_t)(r + 8 * kh) * 256] = v;
  }
}

// ---------------------------------------------------------------------------
// LSTM recurrence: one block, 512 threads = 16 waves. Each wave owns 64 of
// the 1024 gate columns: 4 WMMA tiles of (4-padded-to-16) x 16 over K=256.
// h/c live in LDS; gx = x2@Wih^T + bih + bhh precomputed.
// ---------------------------------------------------------------------------
__global__ __launch_bounds__(512) void lstm_seq(
    const float* __restrict__ gx,   // [256][4][1024]
    const float* __restrict__ Whh,  // [1024][256] row-major
    float* __restrict__ hist) {     // [256][256]  (h of batch row 3 per step)
  __shared__ float hL[NBL * HDIM];
  __shared__ float cL[NBL * HDIM];
  __shared__ float gL[NBL * 4 * HDIM];   // g[bl][j], j<1024, stored bl*1024+j
  const int tid = threadIdx.x;
  for (int p = tid; p < NBL * HDIM; p += 512) { hL[p] = 0.f; cL[p] = 0.f; }
  __syncthreads();

  const int wave = tid >> 5, lane = tid & 31;
  const int col = lane & 15, kh = lane >> 4;
  const int jb = wave * 64;

  for (int t = 0; t < 256; ++t) {
    v8f acc[4];
#pragma unroll
    for (int nt = 0; nt < 4; ++nt) {
      const int j = jb + nt * 16 + col;
#pragma unroll
      for (int r = 0; r < 8; ++r) {
        const float g0 = gx[(size_t)(t * 4 + (r & 3)) * 1024 + j];
        acc[nt][r] = (kh == 0 && r < 4) ? g0 : 0.f;   // rows >= 4 are padding
      }
    }
#pragma unroll 4
    for (int k0 = 0; k0 < 256; k0 += 4) {
      const int rr = col & 3;                         // safe row index
      const v2f av = *(const v2f*)(hL + rr * 256 + k0 + 2 * kh);
      v2f a;                                          // zero-pad rows 4..15
      a.x = (col < 4) ? av.x : 0.f;
      a.y = (col < 4) ? av.y : 0.f;
#pragma unroll
      for (int nt = 0; nt < 4; ++nt) {
        const int j = jb + nt * 16 + col;
        const v2f b = *(const v2f*)(Whh + (size_t)j * 256 + k0 + 2 * kh);
        acc[nt] = __builtin_amdgcn_wmma_f32_16x16x4_f32(false, a, false, b,
                                                        (short)0, acc[nt], false, false);
      }
    }
#pragma unroll
    for (int nt = 0; nt < 4; ++nt) {
      if (kh == 0) {                                  // lanes 0..15 hold M=r (r<8)
        const int j = jb + nt * 16 + col;
#pragma unroll
        for (int r = 0; r < 4; ++r) gL[r * 1024 + j] = acc[nt][r];
      }
    }
    __syncthreads();
    for (int p = tid; p < NBL * HDIM; p += 512) {
      const int bl = p >> 8, jj = p & 255;
      const float gi = gL[bl * 1024 + jj];
      const float gf = gL[bl * 1024 + 256 + jj];
      const float gg = gL[bl * 1024 + 512 + jj];
      const float go = gL[bl * 1024 + 768 + jj];
      float c = sigmoidf(gf) * cL[p] + sigmoidf(gi) * tanhf(gg);
      float h = sigmoidf(go) * tanhf(c);
      cL[p] = c; hL[p] = h;
      if (bl == 3) hist[t * 256 + jj] = h;
    }
    __syncthreads();
  }
}

// ------------------------------- sparse / misc -----------------------------
__global__ void k_deg(const int* __restrict__ dst, const float* __restrict__ w,
                      float* deg, int E) {
  const int e = blockIdx.x * blockDim.x + threadIdx.x;
  if (e >= E) return;
  const float we = w ? w[e] : 1.f;
  if (we > 0.f) fatomic_add(&deg[dst[e]], we);
}

__global__ void k_dinv(const float* deg, float* dinv, int n) {
  const int i = blockIdx.x * blockDim.x + threadIdx.x;
  if (i < n) dinv[i] = rsqrtf(deg[i] + 1.f);
}

__global__ void k_sagg(const int* __restrict__ src, const int* __restrict__ dst,
                       const float* __restrict__ w, const float* __restrict__ dinv,
                       const float* __restrict__ sval, float* sagg, int E) {
  const int e = blockIdx.x * blockDim.x + threadIdx.x;
  if (e >= E) return;
  const float we = w ? w[e] : 1.f;
  if (we <= 0.f) return;
  const int s = src[e], d = dst[e];
  fatomic_add(&sagg[d], dinv[s] * dinv[d] * we * sval[s]);
}

// GCN1 is rank-1: out[i][j] = relu((sagg[i] + dinv^2 x[i]) * W1[j] + b1[j])
__global__ void k_gcn1(const float* __restrict__ x, const float* __restrict__ sagg,
                       const float* __restrict__ dinv, const float* __restrict__ W1,
                       const float* __restrict__ b1, float* __restrict__ h1) {
  const int i = blockIdx.x, j = threadIdx.x;
  const float base = sagg[i] + dinv[i] * dinv[i] * x[i];
  h1[(size_t)i * 256 + j] = fmaxf(base * W1[j] + b1[j], 0.f);
}

// per-node matvec (score projection): one wave per node
__global__ __launch_bounds__(256) void k_matvec(const float* __restrict__ hmat,
                                                const float* __restrict__ w,
                                                float* s, int n) {
  const int g = blockIdx.x * blockDim.x + threadIdx.x;
  const int node = g >> 5, lane = g & 31;
  if (node >= n) return;
  const float* r = hmat + (size_t)node * 256;
  float acc = 0.f;
#pragma unroll
  for (int j = 0; j < 8; ++j) acc += r[lane + j * 32] * w[lane + j * 32];
#pragma unroll
  for (int off = 16; off > 0; off >>= 1) acc += __shfl_down(acc, off, 32);
  if (lane == 0) s[node] = acc;
}

__global__ void k_score(const float* sagg, const float* dinv, const float* sval,
                        const float* sb, float* score, int n) {
  const int i = blockIdx.x * blockDim.x + threadIdx.x;
  if (i < n) score[i] = sagg[i] + dinv[i] * dinv[i] * sval[i] + sb[0];
}

// per-graph descending bitonic sort; blockDim.x == per (512 or 256)
__global__ void k_topk(const float* __restrict__ score, int* __restrict__ perm, int k) {
  __shared__ float key[512];
  __shared__ int   kid[512];
  const int n = blockDim.x, t = threadIdx.x, g = blockIdx.x;
  key[t] = score[g * n + t];
  kid[t] = t;
  __syncthreads();
  for (int size = 2; size <= n; size <<= 1) {
    for (int stride = size >> 1; stride > 0; stride >>= 1) {
      const int ixj = t ^ stride;
      if (ixj > t) {
        const bool desc = ((t & size) == 0);
        const float a = key[t], b = key[ixj];
        if (desc ? (a < b) : (a > b)) {
          key[t] = b; key[ixj] = a;
          const int ti = kid[t]; kid[t] = kid[ixj]; kid[ixj] = ti;
        }
      }
      __syncthreads();
    }
  }
  if (t < k) perm[g * k + t] = g * n + kid[t];
}

__global__ void k_invset(const int* perm, int* inv, int nk) {
  const int i = blockIdx.x * blockDim.x + threadIdx.x;
  if (i < nk) inv[perm[i]] = i;
}

__global__ void k_gather(const float* __restrict__ hmat, const float* __restrict__ score,
                         const int* __restrict__ perm, float* __restrict__ xp) {
  const int i = blockIdx.x, j = threadIdx.x;
  const int old = perm[i];
  xp[(size_t)i * 256 + j] = hmat[(size_t)old * 256 + j] * tanhf(score[old]);
}

__global__ void k_relabel(const int* __restrict__ src, const int* __restrict__ dst,
                          const float* __restrict__ w, const int* __restrict__ inv,
                          int* ns, int* nd, float* nw, int E) {
  const int e = blockIdx.x * blockDim.x + threadIdx.x;
  if (e >= E) return;
  const float we = w ? w[e] : 1.f;
  const int vs = inv[src[e]], vd = inv[dst[e]];
  const bool valid = (we > 0.f) && (vs >= 0) && (vd >= 0);
  ns[e] = valid ? vs : 0;
  nd[e] = valid ? vd : 0;
  nw[e] = valid ? 1.f : 0.f;
}

__global__ void k_selfinit(const float* __restrict__ hg, const float* __restrict__ dinv,
                           const float* __restrict__ b, float* __restrict__ out) {
  const int i = blockIdx.x, j = threadIdx.x;
  out[(size_t)i * 256 + j] = dinv[i] * dinv[i] * hg[(size_t)i * 256 + j] + b[j];
}

// full-feature scatter: one wave per edge, 8 features per lane
__global__ __launch_bounds__(256) void k_scatter(
    const int* __restrict__ src, const int* __restrict__ dst,
    const float* __restrict__ w, const float* __restrict__ dinv,
    const float* __restrict__ hg, float* __restrict__ out, int E) {
  const int g = blockIdx.x * blockDim.x + threadIdx.x;
  const int e = g >> 5, lane = g & 31;
  if (e >= E) return;
  const float we = w ? w[e] : 1.f;
  if (we <= 0.f) return;
  const int s = src[e], d = dst[e];
  const float norm = dinv[s] * dinv[d] * we;
  const float* hr = hg + (size_t)s * 256;
  float* orow = out + (size_t)d * 256;
#pragma unroll
  for (int r = 0; r < 8; ++r) {
    const int j = lane + r * 32;
    fatomic_add(&orow[j], norm * hr[j]);
  }
}

__global__ void k_relu(float* p, size_t n) {
  const size_t i = (size_t)blockIdx.x * blockDim.x + threadIdx.x;
  if (i < n) p[i] = fmaxf(p[i], 0.f);
}

__global__ void k_meanpool(const float* __restrict__ out3, float* __restrict__ x1) {
  const int g = blockIdx.x, j = threadIdx.x;
  float acc = 0.f;
  for (int r = 0; r < KEEP2; ++r) acc += out3[(size_t)(g * KEEP2 + r) * 256 + j];
  x1[g * 256 + j] = acc * (1.f / KEEP2);
}

// gx[t][bl][j] = bih[j] + bhh[j] + sum_k x2[t][bl][k]*Wih[j][k]
__global__ __launch_bounds__(1024) void k_gx(const float* __restrict__ x2,
                                             const float* __restrict__ Wih,
                                             const float* __restrict__ bih,
                                             const float* __restrict__ bhh,
                                             float* __restrict__ gx) {
  const int j = threadIdx.x, tb = blockIdx.x;   // tb = t*4+bl
  const float* xr = x2 + tb * INDIM;
  const float* wr = Wih + j * INDIM;
  float acc = bih[j] + bhh[j];
#pragma unroll
  for (int kk = 0; kk < INDIM; ++kk) acc += xr[kk] * wr[kk];
  gx[(size_t)tb * 1024 + j] = acc;
}

__global__ void k_final(const float* __restrict__ x1, const float* __restrict__ xl,
                        const float* __restrict__ Wo, const float* __restrict__ bo,
                        float* __restrict__ out) {
  const int t = threadIdx.x;                    // 512 threads
  const int b = t >> 1, c = t & 1;
  float acc = bo[c];
  for (int j = 0; j < 256; ++j) acc += x1[b * 256 + j] * Wo[j * 2 + c];
  for (int j = 0; j < 256; ++j) acc += xl[b * 256 + j] * Wo[(256 + j) * 2 + c];
  out[b * 2 + c] = acc;
}

// ---------------------------------------------------------------------------
extern "C" void kernel_launch(void* const* d_in, const int* in_sizes, int n_in,
                              void* d_out, int out_size, void* d_ws, size_t ws_size,
                              hipStream_t stream) {
  (void)in_sizes; (void)n_in; (void)out_size; (void)ws_size;
  const float* x   = (const float*)d_in[0];
  const int*   ei  = (const int*)d_in[1];
  const float* x2  = (const float*)d_in[3];
  const float* W1  = (const float*)d_in[4];
  const float* b1  = (const float*)d_in[5];
  const float* sW1 = (const float*)d_in[6];
  const float* sb1 = (const float*)d_in[7];
  const float* W2  = (const float*)d_in[8];
  const float* b2  = (const float*)d_in[9];
  const float* sW2 = (const float*)d_in[10];
  const float* sb2 = (const float*)d_in[11];
  const float* W3  = (const float*)d_in[12];
  const float* b3  = (const float*)d_in[13];
  const float* Wih = (const float*)d_in[14];
  const float* Whh = (const float*)d_in[15];
  const float* bih = (const float*)d_in[16];
  const float* bhh = (const float*)d_in[17];
  const float* Wl  = (const float*)d_in[18];
  const float* bl  = (const float*)d_in[19];
  const float* Wo  = (const float*)d_in[20];
  const float* bo  = (const float*)d_in[21];
  float* out = (float*)d_out;

  const int* src1 = ei;
  const int* dst1 = ei + NEDGE;

  // ---- workspace layout (~224 MiB, regions reused across stages) ----
  char* ws = (char*)d_ws;
  size_t off = 0;
  auto alloc = [&](size_t bytes) -> char* {
    char* p = ws + off; off += (bytes + 255) & ~(size_t)255; return p;
  };
  float* regA = (float*)alloc((size_t)NODE1 * HDIM * 4);   // 134 MB
  float* regB = (float*)alloc((size_t)NODE2 * HDIM * 4);   //  67 MB
  float* deg  = (float*)alloc((size_t)NODE1 * 4);
  float* dinv = (float*)alloc((size_t)NODE1 * 4);
  float* sval = (float*)alloc((size_t)NODE1 * 4);
  float* sagg = (float*)alloc((size_t)NODE1 * 4);
  float* scor = (float*)alloc((size_t)NODE1 * 4);
  int*   perm = (int*)alloc((size_t)NODE2 * 4);
  int*   inv  = (int*)alloc((size_t)NODE1 * 4);
  int*   src2 = (int*)alloc((size_t)NEDGE * 4);
  int*   dst2 = (int*)alloc((size_t)NEDGE * 4);
  float* w2   = (float*)alloc((size_t)NEDGE * 4);
  int*   src3 = (int*)alloc((size_t)NEDGE * 4);
  int*   dst3 = (int*)alloc((size_t)NEDGE * 4);
  float* w3   = (float*)alloc((size_t)NEDGE * 4);
  float* gx   = (float*)alloc((size_t)256 * 4 * 1024 * 4);
  float* hist = (float*)alloc((size_t)256 * 256 * 4);
  float* x1b  = (float*)alloc((size_t)256 * 256 * 4);
  float* xlb  = (float*)alloc((size_t)256 * 256 * 4);

  // region aliases (lifetimes are disjoint)
  float* h1   = regA;                             // stage1 features
  float* hg2  = regA;                             // GEMM2 result (h1 dead)
  float* out2 = regA + (size_t)NODE2 * HDIM;      // GCN2 output
  float* out3 = regA;                             // GCN3 output (hg2 dead)
  float* hp1  = regB;                             // pooled-1 features
  float* hp2  = regB;                             // pooled-2 features (hp1 dead)
  float* hg3  = regB + (size_t)NODE3 * HDIM;      // GEMM3 result

  const int EB = (NEDGE + 255) / 256;

  // ================= Stage 1: GCN1 (rank-1) + SAGPool1 =================
  hipMemsetAsync(deg, 0, (size_t)NODE1 * 4, stream);
  k_deg<<<EB, 256, 0, stream>>>(dst1, nullptr, deg, NEDGE);
  k_dinv<<<(NODE1 + 255) / 256, 256, 0, stream>>>(deg, dinv, NODE1);
  hipMemsetAsync(sagg, 0, (size_t)NODE1 * 4, stream);
  k_sagg<<<EB, 256, 0, stream>>>(src1, dst1, nullptr, dinv, x, sagg, NEDGE);
  k_gcn1<<<NODE1, 256, 0, stream>>>(x, sagg, dinv, W1, b1, h1);

  k_matvec<<<NODE1 / 8, 256, 0, stream>>>(h1, sW1, sval, NODE1);
  hipMemsetAsync(sagg, 0, (size_t)NODE1 * 4, stream);
  k_sagg<<<EB, 256, 0, stream>>>(src1, dst1, nullptr, dinv, sval, sagg, NEDGE);
  k_score<<<(NODE1 + 255) / 256, 256, 0, stream>>>(sagg, dinv, sval, sb1, scor, NODE1);
  k_topk<<<NGRAPH, PER1, 0, stream>>>(scor, perm, KEEP1);
  hipMemsetAsync(inv, 0xFF, (size_t)NODE1 * 4, stream);                 // -1
  k_invset<<<(NODE2 + 255) / 256, 256, 0, stream>>>(perm, inv, NODE2);
  k_gather<<<NODE2, 256, 0, stream>>>(h1, scor, perm, hp1);
  k_relabel<<<EB, 256, 0, stream>>>(src1, dst1, nullptr, inv, src2, dst2, w2, NEDGE);

  // ================= Stage 2: GCN2 + SAGPool2 =================
  hipMemsetAsync(deg, 0, (size_t)NODE2 * 4, stream);
  k_deg<<<EB, 256, 0, stream>>>(dst2, w2, deg, NEDGE);
  k_dinv<<<(NODE2 + 255) / 256, 256, 0, stream>>>(deg, dinv, NODE2);
  gemm_wmma256<<<dim3(16, NODE2 / 128), 256, 0, stream>>>(hp1, W2, hg2, nullptr, 0);
  k_selfinit<<<NODE2, 256, 0, stream>>>(hg2, dinv, b2, out2);
  k_scatter<<<NEDGE / 8, 256, 0, stream>>>(src2, dst2, w2, dinv, hg2, out2, NEDGE);
  k_relu<<<(int)(((size_t)NODE2 * HDIM) / 256), 256, 0, stream>>>(out2, (size_t)NODE2 * HDIM);

  k_matvec<<<NODE2 / 8, 256, 0, stream>>>(out2, sW2, sval, NODE2);
  hipMemsetAsync(sagg, 0, (size_t)NODE2 * 4, stream);
  k_sagg<<<EB, 256, 0, stream>>>(src2, dst2, w2, dinv, sval, sagg, NEDGE);
  k_score<<<(NODE2 + 255) / 256, 256, 0, stream>>>(sagg, dinv, sval, sb2, scor, NODE2);
  k_topk<<<NGRAPH, PER2, 0, stream>>>(scor, perm, KEEP2);
  hipMemsetAsync(inv, 0xFF, (size_t)NODE2 * 4, stream);
  k_invset<<<(NODE3 + 255) / 256, 256, 0, stream>>>(perm, inv, NODE3);
  k_gather<<<NODE3, 256, 0, stream>>>(out2, scor, perm, hp2);
  k_relabel<<<EB, 256, 0, stream>>>(src2, dst2, w2, inv, src3, dst3, w3, NEDGE);

  // ================= Stage 3: GCN3 (no relu) + mean pool =================
  hipMemsetAsync(deg, 0, (size_t)NODE3 * 4, stream);
  k_deg<<<EB, 256, 0, stream>>>(dst3, w3, deg, NEDGE);
  k_dinv<<<(NODE3 + 255) / 256, 256, 0, stream>>>(deg, dinv, NODE3);
  gemm_wmma256<<<dim3(16, NODE3 / 128), 256, 0, stream>>>(hp2, W3, hg3, nullptr, 0);
  k_selfinit<<<NODE3, 256, 0, stream>>>(hg3, dinv, b3, out3);
  k_scatter<<<NEDGE / 8, 256, 0, stream>>>(src3, dst3, w3, dinv, hg3, out3, NEDGE);
  k_meanpool<<<NGRAPH, 256, 0, stream>>>(out3, x1b);

  // ================= LSTM branch =================
  k_gx<<<256 * NBL, 1024, 0, stream>>>(x2, Wih, bih, bhh, gx);
  lstm_seq<<<1, 512, 0, stream>>>(gx, Whh, hist);
  gemm_wmma256<<<dim3(16, 2), 256, 0, stream>>>(hist, Wl, xlb, bl, 0);

  // ================= classifier =================
  k_final<<<1, 512, 0, stream>>>(x1b, xlb, Wo, bo, out);
}